// CrossAttentionBlock_12773232738807
// MI455X (gfx1250) — compile-verified
//
#include <hip/hip_runtime.h>

typedef __attribute__((ext_vector_type(2))) float v2f;
typedef __attribute__((ext_vector_type(8))) float v8f;

#define LDSS 258  // padded LDS row stride (floats): 2*m mod 64 unique for m<32 -> no bank conflicts

// ---------------------------------------------------------------------------
// Kernel 1: single workgroup (256 threads = 8 waves).
//   a = layer_norm(act) ; v = a@vw + vb ; y = v@ow + ob   (all 32x256, K=256)
// GEMMs run on V_WMMA_F32_16X16X4_F32 (fp32 in, fp32 acc).
// ---------------------------------------------------------------------------
__global__ __launch_bounds__(256) void ca_compute_y(
    const float* __restrict__ act,
    const float* __restrict__ ln_w, const float* __restrict__ ln_b,
    const float* __restrict__ vw,   const float* __restrict__ vb,
    const float* __restrict__ ow,   const float* __restrict__ ob,
    float* __restrict__ y)
{
    __shared__ float smem[32 * LDSS];

    const int tid  = threadIdx.x;
    const int wave = tid >> 5;
    const int lane = tid & 31;

    // ---- LayerNorm(act) -> smem.  Each wave handles 4 rows of 256. ----
    for (int r = wave * 4; r < wave * 4 + 4; ++r) {
        float x[8];
        float s = 0.f, ss = 0.f;
        #pragma unroll
        for (int j = 0; j < 8; ++j) {
            x[j] = act[r * 256 + lane + 32 * j];
            s  += x[j];
            ss += x[j] * x[j];
        }
        #pragma unroll
        for (int msk = 16; msk >= 1; msk >>= 1) {
            s  += __shfl_xor(s,  msk, 32);
            ss += __shfl_xor(ss, msk, 32);
        }
        const float mu  = s * (1.0f / 256.0f);
        const float var = ss * (1.0f / 256.0f) - mu * mu;
        const float rs  = rsqrtf(var + 1e-5f);
        #pragma unroll
        for (int j = 0; j < 8; ++j) {
            const int c = lane + 32 * j;
            smem[r * LDSS + c] = (x[j] - mu) * rs * ln_w[c] + ln_b[c];
        }
    }
    __syncthreads();

    // ---- WMMA fragment coordinates (16x16x4 f32) ----
    const int m  = lane & 15;   // A row / B,C,D column position
    const int kh = lane >> 4;   // K half: lanes 16-31 carry K+2 / rows M+8

    v8f acc[4];

    // ---- GEMM1: v = a_ln @ vw  (M=32 -> 2 tiles, N=256 -> 16 tiles) ----
    #pragma unroll
    for (int i = 0; i < 4; ++i) {
        const int tile = wave * 4 + i;        // 0..31
        const int mt = tile >> 4;             // waves 0-3: mt=0, waves 4-7: mt=1
        const int nt = tile & 15;
        const int nc = nt * 16 + m;
        const float* arow = &smem[(mt * 16 + m) * LDSS];
        v8f c = {};
        for (int k = 0; k < 256; k += 4) {
            v2f a = *(const v2f*)&arow[k + 2 * kh];                 // ds_load_b64
            v2f b;
            b.x = vw[(k + 2 * kh + 0) * 256 + nc];
            b.y = vw[(k + 2 * kh + 1) * 256 + nc];
            c = __builtin_amdgcn_wmma_f32_16x16x4_f32(
                    false, a, false, b, (short)0, c, false, false);
        }
        acc[i] = c;
    }
    __syncthreads();  // everyone done reading a_ln from smem

    // ---- store v = gemm1 + vb into smem (reuse buffer) ----
    #pragma unroll
    for (int i = 0; i < 4; ++i) {
        const int tile = wave * 4 + i;
        const int mt = tile >> 4;
        const int nt = tile & 15;
        const int nc = nt * 16 + m;
        const float bias = vb[nc];
        #pragma unroll
        for (int vr = 0; vr < 8; ++vr) {
            const int row = mt * 16 + vr + 8 * kh;   // C/D layout: M = vr + 8*kh
            smem[row * LDSS + nc] = acc[i][vr] + bias;
        }
    }
    __syncthreads();

    // ---- GEMM2: y = v @ ow + ob -> global workspace ----
    #pragma unroll
    for (int i = 0; i < 4; ++i) {
        const int tile = wave * 4 + i;
        const int mt = tile >> 4;
        const int nt = tile & 15;
        const int nc = nt * 16 + m;
        const float* arow = &smem[(mt * 16 + m) * LDSS];
        v8f c = {};
        for (int k = 0; k < 256; k += 4) {
            v2f a = *(const v2f*)&arow[k + 2 * kh];
            v2f b;
            b.x = ow[(k + 2 * kh + 0) * 256 + nc];
            b.y = ow[(k + 2 * kh + 1) * 256 + nc];
            c = __builtin_amdgcn_wmma_f32_16x16x4_f32(
                    false, a, false, b, (short)0, c, false, false);
        }
        const float bias = ob[nc];
        #pragma unroll
        for (int vr = 0; vr < 8; ++vr) {
            const int row = mt * 16 + vr + 8 * kh;
            y[row * 256 + nc] = c[vr] + bias;
        }
    }
}

// ---------------------------------------------------------------------------
// Kernel 2: out = img + broadcast(y).  Pure HBM-bandwidth streaming kernel.
// Each block covers 1024 consecutive floats of one (b,c) channel, so
// y[blockIdx.x >> 2] is block-uniform -> scalar load.
// ---------------------------------------------------------------------------
__global__ __launch_bounds__(256) void ca_bcast_add(
    const float* __restrict__ img,
    const float* __restrict__ y,
    float* __restrict__ out)
{
    const float yv = y[blockIdx.x >> 2];               // 4 blocks per channel (4096/1024)
    const size_t idx = (size_t)blockIdx.x * blockDim.x + threadIdx.x;  // float4 index
    const float4* img4 = (const float4*)img;
    float4* out4 = (float4*)out;
    float4 v = img4[idx];
    v.x += yv; v.y += yv; v.z += yv; v.w += yv;
    out4[idx] = v;
}

extern "C" void kernel_launch(void* const* d_in, const int* in_sizes, int n_in,
                              void* d_out, int out_size, void* d_ws, size_t ws_size,
                              hipStream_t stream) {
    // setup_inputs order:
    // 0:img 1:act 2:gn_w 3:gn_b 4:ln_w 5:ln_b 6:qw 7:qb 8:kw 9:kb 10:vw 11:vb 12:ow 13:ob
    const float* img  = (const float*)d_in[0];
    const float* act  = (const float*)d_in[1];
    const float* ln_w = (const float*)d_in[4];
    const float* ln_b = (const float*)d_in[5];
    const float* vw   = (const float*)d_in[10];
    const float* vb   = (const float*)d_in[11];
    const float* ow   = (const float*)d_in[12];
    const float* ob   = (const float*)d_in[13];

    float* y   = (float*)d_ws;    // 32*256 floats = 32 KB scratch
    float* out = (float*)d_out;

    ca_compute_y<<<dim3(1), dim3(256), 0, stream>>>(act, ln_w, ln_b, vw, vb, ow, ob, y);

    // 32*256*64*64 floats = 8,388,608 float4 = 32768 blocks * 256 threads
    ca_bcast_add<<<dim3(32768), dim3(256), 0, stream>>>(img, y, out);
}